// TransientGenerator_76811195122338
// MI455X (gfx1250) — compile-verified
//
#include <hip/hip_runtime.h>
#include <cstdint>

#define SR        16000.0f
#define K_T       20
#define T_LEN     1024
#define B_DIM     64
#define M_DIM     512

#define EV_PER_BLOCK 16
#define CHUNKS       (M_DIM / EV_PER_BLOCK)   // 32 -> 2048 scatter blocks
#define BPR_MAX      128                      // blocks per row, max reduce
#define BPR_SCALE    64                       // blocks per row, scale pass

typedef __attribute__((ext_vector_type(4))) unsigned tdm_g0_t;
typedef __attribute__((ext_vector_type(8))) unsigned tdm_g1_t;

// ---------------------------------------------------------------- zero ------
__global__ __launch_bounds__(256) void tg_zero_kernel(float4* __restrict__ out4,
                                                      long n4,
                                                      unsigned* __restrict__ rowmax) {
    long stride = (long)gridDim.x * blockDim.x;
    for (long i = (long)blockIdx.x * blockDim.x + threadIdx.x; i < n4; i += stride)
        out4[i] = make_float4(0.f, 0.f, 0.f, 0.f);
    if (blockIdx.x == 0 && threadIdx.x < B_DIM)
        rowmax[threadIdx.x] = 0u;   // bits of +0.0f
}

// ------------------------------------------------------------- scatter ------
// One block handles EV_PER_BLOCK events of one batch row. The full template
// bank (20 x 1024 f32 = 80 KB) is DMA'd into LDS with a single CDNA5
// TENSOR_LOAD_TO_LDS (Tensor Data Mover, TENSORcnt path) issued by wave 0,
// then each event does guarded global_atomic_add_f32 scatter into its row.
__global__ __launch_bounds__(256) void tg_scatter_kernel(
    const float* __restrict__ timings, const int* __restrict__ ids,
    const float* __restrict__ gains,   const float* __restrict__ templates,
    float* __restrict__ out, int L) {

    __shared__ float tmpl[K_T * T_LEN];          // 80 KB of the 320 KB/WGP LDS

    if (threadIdx.x < 32) {                      // wave 0 only: TDM is wave-level
        const unsigned lds_off = (unsigned)(uintptr_t)tmpl;  // low 32b = LDS byte offset
        const unsigned long long ga = (unsigned long long)(uintptr_t)templates;

        // D# group 0 (ISA 8.3): count=1, lds_addr, global_addr[56:0], type=2
        tdm_g0_t g0;
        g0.x = 1u;                                   // count=1, user mode
        g0.y = lds_off;                              // lds_addr [63:32]
        g0.z = (unsigned)(ga & 0xFFFFFFFFu);         // global_addr[31:0]
        g0.w = (unsigned)((ga >> 32) & 0x01FFFFFFu)  // global_addr[56:32]
             | (2u << 30);                           // type=2 ("image")

        // D# group 1 (ISA 8.4): 2-D tensor == tile: 1024 x 20 elements of 4 B
        tdm_g1_t g1;
        g1.s0 = (2u << 16);          // wg_mask=0 (not in cluster), data_size=2 (4B)
        g1.s1 = (T_LEN & 0xFFFFu) << 16;      // tensor_dim0[15:0]=1024
        g1.s2 = ((T_LEN >> 16) & 0xFFFFu)     // tensor_dim0[31:16]
              | ((K_T & 0xFFFFu) << 16);      // tensor_dim1[15:0]=20
        g1.s3 = ((K_T >> 16) & 0xFFFFu)       // tensor_dim1[31:16]
              | ((T_LEN & 0xFFFFu) << 16);    // tile_dim0=1024
        g1.s4 = (K_T & 0xFFFFu);              // tile_dim1=20, tile_dim2=0
        g1.s5 = T_LEN;                        // tensor_dim0_stride[31:0]=1024
        g1.s6 = 0u;                           // stride0[47:32]=0, stride1[15:0]=0
        g1.s7 = 0u;                           // tensor_dim1_stride rest = 0

        asm volatile("tensor_load_to_lds %0, %1"
                     :
                     : "s"(g0), "s"(g1)
                     : "memory");
        __builtin_amdgcn_s_wait_tensorcnt(0);        // wave 0's TENSORcnt -> 0
    }
    __syncthreads();                                 // publish LDS to all waves

    const int b     = blockIdx.x / CHUNKS;
    const int chunk = blockIdx.x - b * CHUNKS;
    const int m0    = chunk * EV_PER_BLOCK;
    float* rowOut   = out + (size_t)b * (size_t)L;

    for (int e = 0; e < EV_PER_BLOCK; ++e) {
        const int   m  = m0 + e;
        const float tm = timings[b * M_DIM + m];
        const float g  = gains[b * M_DIM + m];
        int         id = ids[b * M_DIM + m];
        const int  pos = (int)floorf(tm * SR);
        // invalid events contribute exactly zero in the reference -> skip.
        // (branch is uniform: every thread in the block sees the same event)
        if (!((g > 0.0f) && (tm > 0.0f) && (pos < L))) continue;
        if (id > K_T - 1) id = K_T - 1;
        const float* trow = &tmpl[id * T_LEN];
        for (int t = threadIdx.x; t < T_LEN; t += 256) {
            const int idx = pos + t;
            if (idx < L)                       // reference drops the tail
                atomicAdd(&rowOut[idx], g * trow[t]);
        }
    }
}

// -------------------------------------------------------------- rowmax ------
__global__ __launch_bounds__(256) void tg_rowmax_kernel(const float* __restrict__ out,
                                                        unsigned* __restrict__ rowmax,
                                                        int L) {
    const int b      = blockIdx.x / BPR_MAX;
    const int seg    = blockIdx.x - b * BPR_MAX;
    const int segLen = (L + BPR_MAX - 1) / BPR_MAX;
    const int start  = seg * segLen;
    const int end    = min(start + segLen, L);
    const float* row = out + (size_t)b * (size_t)L;

    float m = 0.0f;
    for (int i = start + threadIdx.x; i < end; i += 256)
        m = fmaxf(m, fabsf(row[i]));

    __shared__ float red[256];
    red[threadIdx.x] = m;
    __syncthreads();
    for (int s = 128; s > 0; s >>= 1) {
        if (threadIdx.x < s)
            red[threadIdx.x] = fmaxf(red[threadIdx.x], red[threadIdx.x + s]);
        __syncthreads();
    }
    if (threadIdx.x == 0)
        // all values >= 0 -> uint bit-pattern compare == float compare
        atomicMax(&rowmax[b], __float_as_uint(red[0]));
}

// --------------------------------------------------------------- scale ------
__global__ __launch_bounds__(256) void tg_scale_kernel(float4* __restrict__ out4,
                                                       const unsigned* __restrict__ rowmax,
                                                       int L4) {
    const int b      = blockIdx.x / BPR_SCALE;
    const int seg    = blockIdx.x - b * BPR_SCALE;
    const int segLen = (L4 + BPR_SCALE - 1) / BPR_SCALE;
    const int start  = seg * segLen;
    const int end    = min(start + segLen, L4);

    const float mx = __uint_as_float(rowmax[b]);
    const float s  = 1.0f / fmaxf(mx, 1e-8f);

    float4* row = out4 + (size_t)b * (size_t)L4;
    for (int i = start + threadIdx.x; i < end; i += 256) {
        float4 v = row[i];
        v.x *= s; v.y *= s; v.z *= s; v.w *= s;
        row[i] = v;
    }
}

// -------------------------------------------------------------- launch ------
extern "C" void kernel_launch(void* const* d_in, const int* in_sizes, int n_in,
                              void* d_out, int out_size, void* d_ws, size_t ws_size,
                              hipStream_t stream) {
    const float* timings   = (const float*)d_in[0];
    const int*   ids       = (const int*)d_in[1];
    const float* gains     = (const float*)d_in[2];
    const float* templates = (const float*)d_in[3];
    // d_in[4] is audio_length on device; recover it host-side instead:
    const int  L  = out_size / B_DIM;      // 480000
    const long n4 = (long)out_size / 4;

    float*    out    = (float*)d_out;
    unsigned* rowmax = (unsigned*)d_ws;    // 64 x u32 scratch

    tg_zero_kernel   <<<4096,              256, 0, stream>>>((float4*)out, n4, rowmax);
    tg_scatter_kernel<<<B_DIM * CHUNKS,    256, 0, stream>>>(timings, ids, gains,
                                                             templates, out, L);
    tg_rowmax_kernel <<<B_DIM * BPR_MAX,   256, 0, stream>>>(out, rowmax, L);
    tg_scale_kernel  <<<B_DIM * BPR_SCALE, 256, 0, stream>>>((float4*)out, rowmax, L / 4);
}